// EuclideanDistLoss_25297357373517
// MI455X (gfx1250) — compile-verified
//
#include <hip/hip_runtime.h>

// CDNA5 / gfx1250, wave32. Memory-bound streaming reduction:
// 512 MiB @ 23.3 TB/s => ~22.5 us floor. WMMA (V_WMMA_F32_16X16X4_F32) is used
// as a cross-lane reduction engine so the hot loop has zero DS/shuffle traffic.
typedef __attribute__((ext_vector_type(2))) float v2f;
typedef __attribute__((ext_vector_type(4))) float v4f;
typedef __attribute__((ext_vector_type(8))) float v8f;

#define CUTOFF_F   0.1f
#define PRESSURE_F 10.0f
#define D_FEAT     256          // feature dim (floats per row)
#define BLOCK      256          // 8 wave32s per block
#define WAVES      (BLOCK / 32)
#define ROWS_PER_WAVE 32        // 8 iterations x 4 rows

__device__ __forceinline__ v4f ntload(const float* p) {
  return __builtin_nontemporal_load((const v4f*)p);
}

// Per-lane partial: sum of 8 squared diffs (two float4 pairs).
__device__ __forceinline__ float sqdiff8(v4f a0, v4f a1, v4f b0, v4f b1) {
  float s = 0.0f, t;
  t = a0.x - b0.x; s = fmaf(t, t, s);
  t = a0.y - b0.y; s = fmaf(t, t, s);
  t = a0.z - b0.z; s = fmaf(t, t, s);
  t = a0.w - b0.w; s = fmaf(t, t, s);
  t = a1.x - b1.x; s = fmaf(t, t, s);
  t = a1.y - b1.y; s = fmaf(t, t, s);
  t = a1.z - b1.z; s = fmaf(t, t, s);
  t = a1.w - b1.w; s = fmaf(t, t, s);
  return s;
}

__device__ __forceinline__ float sum8(v8f d) {
  return ((d[0] + d[1]) + (d[2] + d[3])) + ((d[4] + d[5]) + (d[6] + d[7]));
}

__device__ __forceinline__ v8f wmma4(v2f a, v2f b, v8f c) {
  return __builtin_amdgcn_wmma_f32_16x16x4_f32(
      /*neg_a=*/false, a, /*neg_b=*/false, b,
      /*c_mod=*/(short)0, c, /*reuse_a=*/false, /*reuse_b=*/false);
}

// Stage 1: each wave streams 32 contiguous rows, 4 per iteration.
// WMMA ladder per iteration (3 x V_WMMA_F32_16X16X4_F32):
//   D1 = {s_row0,s_row1} x Bsel  -> half-lane row sums of rows 0,1
//   D2 = {s_row2,s_row3} x Bsel  -> half-lane row sums of rows 2,3
//   D3 = {sum8(D1),sum8(D2)} x Bsel -> D3[0] holds full d^2: lanes 0-7 row0,
//        8-15 row2, 16-23 row1, 24-31 row3 (each replicated on 8 lanes).
// Each row's penalty is therefore counted 8x; stage 2 divides it out.
__global__ __launch_bounds__(BLOCK)
void euclid_penalty_stage1(const float* __restrict__ l1,
                           const float* __restrict__ l2,
                           float* __restrict__ partials) {
  __shared__ float waveSum[WAVES];

  const int lane = threadIdx.x & 31;
  const int wave = threadIdx.x >> 5;
  const long long row0 =
      ((long long)blockIdx.x * WAVES + wave) * ROWS_PER_WAVE;

  // Column selector B: cols 0-7 take K={0,2}, cols 8-15 take K={1,3}.
  v2f Bsel;
  Bsel.x = (lane & 8) ? 0.0f : 1.0f;
  Bsel.y = 1.0f - Bsel.x;
  v2f Bones; Bones.x = 1.0f; Bones.y = 1.0f;
  v8f Cz = {};

  const float* base1 = l1 + row0 * D_FEAT + lane * 4;
  const float* base2 = l2 + row0 * D_FEAT + lane * 4;

  float accPen = 0.0f;

  for (int it = 0; it < ROWS_PER_WAVE; it += 4) {
    const float* p1 = base1 + it * D_FEAT;
    const float* p2 = base2 + it * D_FEAT;

    // 16 coalesced b128 loads (512B/wave each), NT hint: pure stream, no reuse.
    v4f r0a0 = ntload(p1 + 0 * D_FEAT), r0a1 = ntload(p1 + 0 * D_FEAT + 128);
    v4f r1a0 = ntload(p1 + 1 * D_FEAT), r1a1 = ntload(p1 + 1 * D_FEAT + 128);
    v4f r2a0 = ntload(p1 + 2 * D_FEAT), r2a1 = ntload(p1 + 2 * D_FEAT + 128);
    v4f r3a0 = ntload(p1 + 3 * D_FEAT), r3a1 = ntload(p1 + 3 * D_FEAT + 128);
    v4f r0b0 = ntload(p2 + 0 * D_FEAT), r0b1 = ntload(p2 + 0 * D_FEAT + 128);
    v4f r1b0 = ntload(p2 + 1 * D_FEAT), r1b1 = ntload(p2 + 1 * D_FEAT + 128);
    v4f r2b0 = ntload(p2 + 2 * D_FEAT), r2b1 = ntload(p2 + 2 * D_FEAT + 128);
    v4f r3b0 = ntload(p2 + 3 * D_FEAT), r3b1 = ntload(p2 + 3 * D_FEAT + 128);

    const float s0 = sqdiff8(r0a0, r0a1, r0b0, r0b1);
    const float s1 = sqdiff8(r1a0, r1a1, r1b0, r1b1);
    const float s2 = sqdiff8(r2a0, r2a1, r2b0, r2b1);
    const float s3 = sqdiff8(r3a0, r3a1, r3b0, r3b1);

    v2f A1; A1.x = s0; A1.y = s1;
    v2f A2; A2.x = s2; A2.y = s3;
    v8f D1 = wmma4(A1, Bsel, Cz);
    v8f D2 = wmma4(A2, Bsel, Cz);

    v2f A3; A3.x = sum8(D1); A3.y = sum8(D2);
    v8f D3 = wmma4(A3, Bsel, Cz);

    const float d2  = D3[0];                       // this lane's row d^2
    const float d   = __builtin_amdgcn_sqrtf(d2);  // raw v_sqrt_f32 (args ~1e2)
    const float dev = d - CUTOFF_F;
    float pen = dev * dev;
    pen = (dev > 0.0f) ? pen : pen * PRESSURE_F;
    accPen += pen;                                 // all lanes active
  }

  // Wave-total of accPen via the same two-WMMA ladder (no DS traffic).
  v2f Ar; Ar.x = accPen; Ar.y = 0.0f;
  v8f Dr1 = wmma4(Ar, Bones, Cz);
  v2f Ar2; Ar2.x = sum8(Dr1); Ar2.y = 0.0f;
  v8f Dr2 = wmma4(Ar2, Bones, Cz);

  if (lane == 0) waveSum[wave] = Dr2[0];
  __syncthreads();

  if (threadIdx.x == 0) {
    float s = 0.0f;
    for (int i = 0; i < WAVES; ++i) s += waveSum[i];  // fixed order
    partials[blockIdx.x] = s;
  }
}

// Stage 2: deterministic tree reduction of block partials -> mean.
__global__ __launch_bounds__(256)
void euclid_penalty_stage2(const float* __restrict__ partials, int n,
                           float* __restrict__ out, float scale) {
  __shared__ float sdata[256];
  float s = 0.0f;
  for (int i = threadIdx.x; i < n; i += 256) s += partials[i];
  sdata[threadIdx.x] = s;
  __syncthreads();
  for (int stride = 128; stride; stride >>= 1) {
    if ((int)threadIdx.x < stride) sdata[threadIdx.x] += sdata[threadIdx.x + stride];
    __syncthreads();
  }
  if (threadIdx.x == 0) out[0] = sdata[0] * scale;
}

extern "C" void kernel_launch(void* const* d_in, const int* in_sizes, int n_in,
                              void* d_out, int out_size, void* d_ws, size_t ws_size,
                              hipStream_t stream) {
  const float* l1 = (const float*)d_in[0];
  const float* l2 = (const float*)d_in[1];
  float* out = (float*)d_out;
  float* ws  = (float*)d_ws;

  const long long elems = (long long)in_sizes[0];   // 262144 * 256
  const long long rows  = elems / D_FEAT;           // 262144
  const int grid = (int)(rows / ((long long)WAVES * ROWS_PER_WAVE));  // 1024

  euclid_penalty_stage1<<<grid, BLOCK, 0, stream>>>(l1, l2, ws);

  // Each row's penalty is counted 8x in the partials.
  const float scale = 1.0f / (8.0f * (float)rows);
  euclid_penalty_stage2<<<1, 256, 0, stream>>>(ws, grid, out, scale);
}